// VectorNet_prediction_61555471286756
// MI455X (gfx1250) — compile-verified
//
#include <hip/hip_runtime.h>
#include <hip/hip_bf16.h>

typedef __attribute__((ext_vector_type(2))) float v2f;
typedef __attribute__((ext_vector_type(8))) float v8f;

#define NV 50     // vectors per polyline
#define GD 48     // final feature width (G)
#define SEQ 1024  // B*P
#define EPSF 1e-6f

// ---------------------------------------------------------------------------
// Stage 1: polyline subgraph. One block per (b,p). All FP32 VALU (d<=48).
// Uses the rank-1 structure of the pairwise einsum:
//   agg[i,:] = S*(a[i,:] + be) + (sum_j m_j h[j,:]) @ we_b,  S = sum_j m_j
// ---------------------------------------------------------------------------
struct LayerP {
    const float *w1, *b1, *w2, *b2, *be, *wea, *web, *wp1, *bp1, *wp2, *bp2;
};
struct SubP { LayerP l[3]; };

__device__ inline void dense50(const float* __restrict__ src, float* __restrict__ dst,
                               const float* __restrict__ W, const float* __restrict__ b,
                               int din, int dout, bool relu, const float* __restrict__ mrow) {
    for (int idx = threadIdx.x; idx < NV * dout; idx += blockDim.x) {
        int i = idx / dout, o = idx % dout;
        float s = b ? b[o] : 0.f;
        const float* sr = src + i * GD;
        for (int k = 0; k < din; ++k) s += sr[k] * W[k * dout + o];
        if (relu) s = fmaxf(s, 0.f);
        if (mrow) s *= mrow[i];
        dst[i * GD + o] = s;
    }
}

__global__ void subgraph_kernel(const float* __restrict__ poly_in,
                                const float* __restrict__ masks,
                                SubP P, float* __restrict__ poly_out) {
    __shared__ float buf0[NV * GD];
    __shared__ float buf1[NV * GD];
    __shared__ float mbuf[NV];
    __shared__ float hm[24];
    __shared__ float cm[GD];
    __shared__ float Ssum;

    const int bp  = blockIdx.x;
    const int tid = threadIdx.x;

    for (int idx = tid; idx < NV * 6; idx += blockDim.x)
        buf0[(idx / 6) * GD + (idx % 6)] = poly_in[bp * (NV * 6) + idx];
    if (tid < NV) mbuf[tid] = masks[bp * NV + tid];
    __syncthreads();
    if (tid == 0) {
        float s = 0.f;
        for (int j = 0; j < NV; ++j) s += mbuf[j];
        Ssum = s;
    }
    __syncthreads();

    float* xb = buf0;
    float* hb = buf1;
    for (int layer = 0; layer < 3; ++layer) {
        const LayerP& L = P.l[layer];
        const int d = 6 << layer, dd = d * 2;
        // t = relu(x@w1+b1)
        dense50(xb, hb, L.w1, L.b1, d, d, true, nullptr);  __syncthreads();
        // h = t@w2+b2
        dense50(hb, xb, L.w2, L.b2, d, d, false, nullptr); __syncthreads();
        // a = h@we_a
        dense50(xb, hb, L.wea, nullptr, d, dd, false, nullptr);
        // hm[k] = sum_j m_j * h[j,k]
        if (tid < d) {
            float s = 0.f;
            for (int j = 0; j < NV; ++j) s += mbuf[j] * xb[j * GD + tid];
            hm[tid] = s;
        }
        __syncthreads();
        // cm[o] = hm @ we_b
        if (tid < dd) {
            float s = 0.f;
            for (int k = 0; k < d; ++k) s += hm[k] * L.web[k * dd + tid];
            cm[tid] = s;
        }
        __syncthreads();
        // agg = S*(a+be)+cm, in place on hb
        for (int idx = tid; idx < NV * dd; idx += blockDim.x) {
            int i = idx / dd, o = idx % dd;
            hb[i * GD + o] = Ssum * (hb[i * GD + o] + L.be[o]) + cm[o];
        }
        __syncthreads();
        dense50(hb, xb, L.wp1, L.bp1, dd, dd, true, nullptr);  __syncthreads();
        dense50(xb, hb, L.wp2, L.bp2, dd, dd, false, mbuf);    __syncthreads();
        float* t = xb; xb = hb; hb = t;   // new x is in hb
    }
    if (tid < GD) {
        float s = 0.f;
        for (int i = 0; i < NV; ++i) s += xb[i * GD + tid];
        poly_out[bp * GD + tid] = s / (Ssum + EPSF);
    }
}

// ---------------------------------------------------------------------------
// Stage 2: FP32 WMMA GEMM:  C[M x N] = act(A[M x K] @ W[K x N] + bias)
// Block = 128 threads (4 waves); blockIdx.x = 64-row M super-tile (one 16-row
// tile per wave), blockIdx.y = 16-col N tile. The K x 16 weight slice + bias
// are staged into LDS once with zero-fill for col >= N, so the WMMA K-loop
// has NO predicated loads (previous version compiled the col<N guard into
// s_and_saveexec around every B-element load). K must be a multiple of 4.
// Fragment layouts per CDNA5 ISA 7.12.2 (V_WMMA_F32_16X16X4_F32).
// ---------------------------------------------------------------------------
#define KMAX 64
__global__ void gemm_wmma(const float* __restrict__ A, int lda,
                          const float* __restrict__ W,
                          const float* __restrict__ bias,
                          float* __restrict__ C, int ldc,
                          int K, int N, int relu) {
    __shared__ float wtile[KMAX * 16];  // row-major [k][c]
    __shared__ float btile[16];

    const int n0  = blockIdx.y * 16;
    const int tid = threadIdx.x;

    for (int idx = tid; idx < K * 16; idx += blockDim.x) {
        int kk = idx >> 4, c = idx & 15;
        int col = n0 + c;
        wtile[idx] = (col < N) ? W[kk * N + col] : 0.f;
    }
    if (tid < 16) btile[tid] = (n0 + tid < N) ? bias[n0 + tid] : 0.f;
    __syncthreads();

    const int wave = tid >> 5;
    const int lane = tid & 31;          // wave32
    const int half = lane >> 4;         // K sub-pair select
    const int l    = lane & 15;
    const int m0   = blockIdx.x * 64 + wave * 16;

    v8f acc = {};
    for (int kb = 0; kb < K; kb += 4) {
        v2f a, b;
        const float* ar = A + (m0 + l) * lda + kb + 2 * half;
        a.x = ar[0];
        a.y = ar[1];
        const float* wr = wtile + (kb + 2 * half) * 16 + l;
        b.x = wr[0];
        b.y = wr[16];
        acc = __builtin_amdgcn_wmma_f32_16x16x4_f32(
            /*neg_a=*/false, a, /*neg_b=*/false, b,
            /*c_mod=*/(short)0, acc, /*reuse_a=*/false, /*reuse_b=*/false);
    }
    const int col = n0 + l;
    if (col < N) {
        const float bv = btile[l];
        for (int v = 0; v < 8; ++v) {
            float x = acc[v] + bv;
            if (relu) x = fmaxf(x, 0.f);
            C[(m0 + v + 8 * half) * ldc + col] = x;
        }
    }
}

// ---------------------------------------------------------------------------
// Stage 3: attention over all SEQ tokens (matches reference: cross-batch).
// grid = (SEQ/64 q-tiles, H heads), 256 threads. K/V of the head in LDS.
// hd = 6 -> VALU dot products; 4 threads per query, two-pass softmax + combine.
// ---------------------------------------------------------------------------
__global__ void attn_kernel(const float* __restrict__ Q,
                            const float* __restrict__ Km,
                            const float* __restrict__ Vm,
                            float* __restrict__ att) {
    __shared__ float kb[SEQ * 6];
    __shared__ float vb[SEQ * 6];
    __shared__ float rm[64 * 4];
    __shared__ float rs[64 * 4];
    __shared__ float ra[64 * 4 * 6];

    const int head = blockIdx.y;
    const int tid  = threadIdx.x;

    for (int idx = tid; idx < SEQ * 6; idx += 256) {
        int t = idx / 6, d = idx % 6;
        kb[idx] = Km[t * GD + head * 6 + d];
        vb[idx] = Vm[t * GD + head * 6 + d];
    }
    __syncthreads();

    const int ql   = tid >> 2;
    const int part = tid & 3;
    const int qrow = blockIdx.x * 64 + ql;
    float qv[6];
    for (int d = 0; d < 6; ++d) qv[d] = Q[qrow * GD + head * 6 + d];
    const float scale = 0.4082482904638630f; // 1/sqrt(6)

    const int t0 = part * 256, t1 = t0 + 256;
    float mx = -3.4e38f;
    for (int t = t0; t < t1; ++t) {
        float s = 0.f;
        for (int d = 0; d < 6; ++d) s += qv[d] * kb[t * 6 + d];
        mx = fmaxf(mx, s * scale);
    }
    float sum = 0.f, acc[6] = {0.f, 0.f, 0.f, 0.f, 0.f, 0.f};
    for (int t = t0; t < t1; ++t) {
        float s = 0.f;
        for (int d = 0; d < 6; ++d) s += qv[d] * kb[t * 6 + d];
        float e = __expf(s * scale - mx);
        sum += e;
        for (int d = 0; d < 6; ++d) acc[d] += e * vb[t * 6 + d];
    }
    rm[tid] = mx; rs[tid] = sum;
    for (int d = 0; d < 6; ++d) ra[tid * 6 + d] = acc[d];
    __syncthreads();

    if (part == 0) {
        float M = rm[tid];
        for (int p = 1; p < 4; ++p) M = fmaxf(M, rm[tid + p]);
        float S = 0.f, a[6] = {0.f, 0.f, 0.f, 0.f, 0.f, 0.f};
        for (int p = 0; p < 4; ++p) {
            float w = __expf(rm[tid + p] - M);
            S += rs[tid + p] * w;
            for (int d = 0; d < 6; ++d) a[d] += ra[(tid + p) * 6 + d] * w;
        }
        for (int d = 0; d < 6; ++d) att[qrow * GD + head * 6 + d] = a[d] / S;
    }
}

// ---------------------------------------------------------------------------
// Launch. Input flattening assumed in setup_inputs insertion order:
// 0 polylines, 1 masks, 2 target_index (unused by reference),
// 3..35  subgraph layers x3: w1,b1,w2,b2,be,we_a,we_b,wp1,bp1,wp2,bp2
// 36..49 global: wq,bq,wk,bk,wv,bv,wo,bo,wg1,bg1,wg2,bg2,wd,bd
// 50..53 traj: wt1,bt1,wt2,bt2
// Workspace use: ~2.0 MB of f32 scratch.
// ---------------------------------------------------------------------------
extern "C" void kernel_launch(void* const* d_in, const int* in_sizes, int n_in,
                              void* d_out, int out_size, void* d_ws, size_t ws_size,
                              hipStream_t stream) {
    (void)in_sizes; (void)n_in; (void)out_size; (void)ws_size;
    const float* polylines = (const float*)d_in[0];
    const float* masks     = (const float*)d_in[1];

    SubP P;
    for (int i = 0; i < 3; ++i) {
        int b = 3 + i * 11;
        P.l[i].w1  = (const float*)d_in[b + 0];
        P.l[i].b1  = (const float*)d_in[b + 1];
        P.l[i].w2  = (const float*)d_in[b + 2];
        P.l[i].b2  = (const float*)d_in[b + 3];
        P.l[i].be  = (const float*)d_in[b + 4];
        P.l[i].wea = (const float*)d_in[b + 5];
        P.l[i].web = (const float*)d_in[b + 6];
        P.l[i].wp1 = (const float*)d_in[b + 7];
        P.l[i].bp1 = (const float*)d_in[b + 8];
        P.l[i].wp2 = (const float*)d_in[b + 9];
        P.l[i].bp2 = (const float*)d_in[b + 10];
    }
    const float *wq = (const float*)d_in[36], *bq = (const float*)d_in[37];
    const float *wk = (const float*)d_in[38], *bk = (const float*)d_in[39];
    const float *wv = (const float*)d_in[40], *bv = (const float*)d_in[41];
    const float *wo = (const float*)d_in[42], *bo = (const float*)d_in[43];
    const float *wg1 = (const float*)d_in[44], *bg1 = (const float*)d_in[45];
    const float *wg2 = (const float*)d_in[46], *bg2 = (const float*)d_in[47];
    const float *wd = (const float*)d_in[48], *bd = (const float*)d_in[49];
    const float *wt1 = (const float*)d_in[50], *bt1 = (const float*)d_in[51];
    const float *wt2 = (const float*)d_in[52], *bt2 = (const float*)d_in[53];

    float* ws = (float*)d_ws;
    const size_t SZ = SEQ * GD;         // 49152 floats
    float* poly = ws + 0 * SZ;
    float* q    = ws + 1 * SZ;
    float* k    = ws + 2 * SZ;
    float* v    = ws + 3 * SZ;
    float* att  = ws + 4 * SZ;
    float* atto = ws + 5 * SZ;
    float* g1   = ws + 6 * SZ;
    float* glob = ws + 7 * SZ;
    float* dec  = ws + 8 * SZ;
    float* t1h  = ws + 9 * SZ;          // 1024 x 60
    float* out  = (float*)d_out;        // 1024 x 60

    // 1. subgraph -> poly (1024 x 48)
    subgraph_kernel<<<SEQ, 256, 0, stream>>>(polylines, masks, P, poly);

    const dim3 g48(SEQ / 64, 3);   // N = 48
    const dim3 g60(SEQ / 64, 4);   // N = 60 (zero-padded in LDS)

    // 2. q/k/v projections (WMMA)
    gemm_wmma<<<g48, 128, 0, stream>>>(poly, GD, wq, bq, q, GD, GD, GD, 0);
    gemm_wmma<<<g48, 128, 0, stream>>>(poly, GD, wk, bk, k, GD, GD, GD, 0);
    gemm_wmma<<<g48, 128, 0, stream>>>(poly, GD, wv, bv, v, GD, GD, GD, 0);

    // 3. attention -> att
    attn_kernel<<<dim3(SEQ / 64, 8), 256, 0, stream>>>(q, k, v, att);

    // 4. output proj + FFN + decoder + trajectory head (WMMA)
    gemm_wmma<<<g48, 128, 0, stream>>>(att,  GD, wo,  bo,  atto, GD, GD, GD, 0);
    gemm_wmma<<<g48, 128, 0, stream>>>(atto, GD, wg1, bg1, g1,   GD, GD, GD, 1);
    gemm_wmma<<<g48, 128, 0, stream>>>(g1,   GD, wg2, bg2, glob, GD, GD, GD, 0);
    gemm_wmma<<<g48, 128, 0, stream>>>(glob, GD, wd,  bd,  dec,  GD, GD, GD, 0);
    gemm_wmma<<<g60, 128, 0, stream>>>(dec,  GD, wt1, bt1, t1h,  60, GD, 60, 1);
    gemm_wmma<<<g60, 128, 0, stream>>>(t1h,  60, wt2, bt2, out,  60, 60, 60, 0);
}